// AttentionAggregator_e_2551210574179
// MI455X (gfx1250) — compile-verified
//
#include <hip/hip_runtime.h>
#include <hip/hip_bf16.h>
#include <math.h>

typedef float v2f __attribute__((ext_vector_type(2)));
typedef float v8f __attribute__((ext_vector_type(8)));

#define IN_DIM 256
#define OUT_DIM 128

// ---------------------------------------------------------------------------
// Zero workspace accumulators (num followed contiguously by den)
// ---------------------------------------------------------------------------
__global__ void gat_zero(float* __restrict__ p, size_t count) {
    size_t i = (size_t)blockIdx.x * blockDim.x + threadIdx.x;
    size_t stride = (size_t)gridDim.x * blockDim.x;
    for (; i < count; i += stride) p[i] = 0.0f;
}

// ---------------------------------------------------------------------------
// emb[n][o] = sum_k feats[n][k] * W[o][k] + b[o]
// Full fp32 matrix core: V_WMMA_F32_16X16X4_F32, wave32.
// Block = 256 threads = 8 waves. Each wave computes one 16x16 output tile;
// the 8 waves together cover all OUT_DIM=128 columns for 16 rows.
// A-frag (16x4 f32, 2 VGPRs/lane): lanes 0-15 hold K={0,1}, lanes 16-31 K={2,3}.
// B-frag (4x16 f32):               lanes 0-15 hold K={0,1}, lanes 16-31 K={2,3},
//                                  N = lane&15.  B[k][n] = W[n][k].
// C/D (16x16 f32, 8 VGPRs): VGPR r -> row r (lanes 0-15) / row r+8 (lanes 16-31).
// Epilogue: uniform full-tile check so the common path is 8 plain b32 stores
// (no per-lane EXEC masking); the ragged-tail path never triggers for N=50000.
// ---------------------------------------------------------------------------
__global__ __launch_bounds__(256) void gat_gemm_wmma(
    const float* __restrict__ feats, const float* __restrict__ W,
    const float* __restrict__ bias, float* __restrict__ emb, int n_rows)
{
    const int lane  = threadIdx.x & 31;
    const int wave  = threadIdx.x >> 5;
    const int l15   = lane & 15;
    const int lhalf = lane >> 4;           // 0 -> K pair {0,1}, 1 -> K pair {2,3}

    const int row_base = blockIdx.x * 16;
    const int col_base = wave * 16;        // 8 waves cover 128 output columns

    // Clamp row for loads so EXEC stays all-ones through the WMMA loop
    // (N = 50000 is divisible by 16, so this is just a safety net).
    int arow_idx = row_base + l15;
    if (arow_idx >= n_rows) arow_idx = n_rows - 1;

    const float* arow = feats + (size_t)arow_idx * IN_DIM + lhalf * 2;
    const float* brow = W     + (size_t)(col_base + l15) * IN_DIM + lhalf * 2;

    v8f c = {};
    #pragma unroll 8
    for (int k = 0; k < IN_DIM; k += 4) {
        v2f afrag = *(const v2f*)(arow + k);
        v2f bfrag = *(const v2f*)(brow + k);
        // 8 args: (neg_a, A, neg_b, B, c_mod, C, reuse_a, reuse_b)
        c = __builtin_amdgcn_wmma_f32_16x16x4_f32(
                false, afrag, false, bfrag, (short)0, c, false, false);
    }

    const float bcol = bias[col_base + l15];
    const int out_row_base = row_base + lhalf * 8;   // rows r / r+8 split
    float* orow = emb + (size_t)out_row_base * OUT_DIM + col_base + l15;

    if (row_base + 16 <= n_rows) {
        // Full tile (always taken for N % 16 == 0): straight-line stores.
        #pragma unroll
        for (int r = 0; r < 8; ++r) {
            orow[(size_t)r * OUT_DIM] = c[r] + bcol;
        }
    } else {
        // Ragged tail fallback (never taken for N = 50000).
        #pragma unroll
        for (int r = 0; r < 8; ++r) {
            if (out_row_base + r < n_rows) {
                orow[(size_t)r * OUT_DIM] = c[r] + bcol;
            }
        }
    }
}

// ---------------------------------------------------------------------------
// s_dst[n] = emb[n] . a[0:128],  s_src[n] = emb[n] . a[128:256]
// One wave32 per node; 4 channels per lane; shuffle reduction.
// ---------------------------------------------------------------------------
__global__ __launch_bounds__(256) void gat_scores(
    const float* __restrict__ emb, const float* __restrict__ a,
    float* __restrict__ s_dst, float* __restrict__ s_src, int n)
{
    const int lane = threadIdx.x & 31;
    const int wave = threadIdx.x >> 5;
    const int node = blockIdx.x * 8 + wave;
    if (node >= n) return;

    const float4 e  = *(const float4*)(emb + (size_t)node * OUT_DIM + lane * 4);
    const float4 ad = *(const float4*)(a + lane * 4);
    const float4 as = *(const float4*)(a + OUT_DIM + lane * 4);

    float vd = e.x * ad.x + e.y * ad.y + e.z * ad.z + e.w * ad.w;
    float vs = e.x * as.x + e.y * as.y + e.z * as.z + e.w * as.w;

    #pragma unroll
    for (int off = 16; off > 0; off >>= 1) {
        vd += __shfl_down(vd, off, 32);
        vs += __shfl_down(vs, off, 32);
    }
    if (lane == 0) {
        s_dst[node] = vd;
        s_src[node] = vs;
    }
}

// ---------------------------------------------------------------------------
// Edge aggregation: score = exp(leaky_relu(s_dst[dst] + s_src[src]))
//   num[dst][:] += score * emb[src][:]   (128 fp32 atomics, coalesced)
//   den[dst]    += score
// Block = 128 threads = one thread per channel; EPB edges per block.
// num fits in L2 (25.6 MB of 192 MB) so atomics resolve in-cache.
// ---------------------------------------------------------------------------
#define EPB 4
__global__ __launch_bounds__(128) void gat_edges(
    const int* __restrict__ edges, const float* __restrict__ emb,
    const float* __restrict__ s_dst, const float* __restrict__ s_src,
    float* __restrict__ num, float* __restrict__ den, int n_edges)
{
    const int t  = threadIdx.x;            // output channel
    const int e0 = blockIdx.x * EPB;

    #pragma unroll
    for (int i = 0; i < EPB; ++i) {
        const int e = e0 + i;
        if (e >= n_edges) return;
        const int dst = edges[2 * e];
        const int src = edges[2 * e + 1];
        const float x  = s_dst[dst] + s_src[src];
        const float lr = (x > 0.0f) ? x : 0.1f * x;
        const float sc = expf(lr);
        const float v  = emb[(size_t)src * OUT_DIM + t];
        __hip_atomic_fetch_add(&num[(size_t)dst * OUT_DIM + t], sc * v,
                               __ATOMIC_RELAXED, __HIP_MEMORY_SCOPE_AGENT);
        if (t == 0) {
            __hip_atomic_fetch_add(&den[dst], sc,
                                   __ATOMIC_RELAXED, __HIP_MEMORY_SCOPE_AGENT);
        }
    }
}

// ---------------------------------------------------------------------------
// out[i][:] = num[node_idx[i]][:] / den[node_idx[i]]
// ---------------------------------------------------------------------------
__global__ __launch_bounds__(128) void gat_out(
    const int* __restrict__ node_idx, const float* __restrict__ num,
    const float* __restrict__ den, float* __restrict__ out, int nb)
{
    const int i = blockIdx.x;
    if (i >= nb) return;
    const int t   = threadIdx.x;
    const int nid = node_idx[i];
    out[(size_t)i * OUT_DIM + t] = num[(size_t)nid * OUT_DIM + t] / den[nid];
}

// ---------------------------------------------------------------------------
extern "C" void kernel_launch(void* const* d_in, const int* in_sizes, int n_in,
                              void* d_out, int out_size, void* d_ws, size_t ws_size,
                              hipStream_t stream)
{
    const float* feats    = (const float*)d_in[0];
    const float* W        = (const float*)d_in[1];
    const float* b        = (const float*)d_in[2];
    const float* a        = (const float*)d_in[3];
    const int*   edges    = (const int*)d_in[4];
    const int*   node_idx = (const int*)d_in[5];

    const int N  = in_sizes[0] / IN_DIM;   // 50000
    const int E  = in_sizes[4] / 2;        // 1,610,000
    const int NB = in_sizes[5];            // 10000

    // Workspace layout (floats): emb | num | den | s_dst | s_src
    float* emb = (float*)d_ws;
    float* num = emb + (size_t)N * OUT_DIM;
    float* den = num + (size_t)N * OUT_DIM;
    float* sd  = den + N;
    float* ss  = sd + N;

    // 1. zero num + den (contiguous)
    const size_t zcount = (size_t)N * OUT_DIM + (size_t)N;
    gat_zero<<<2048, 256, 0, stream>>>(num, zcount);

    // 2. emb = feats @ W.T + b   (fp32 WMMA)
    gat_gemm_wmma<<<(N + 15) / 16, 256, 0, stream>>>(feats, W, b, emb, N);

    // 3. attention logits
    gat_scores<<<(N + 7) / 8, 256, 0, stream>>>(emb, a, sd, ss, N);

    // 4. edge aggregation (atomics)
    gat_edges<<<(E + EPB - 1) / EPB, 128, 0, stream>>>(edges, emb, sd, ss, num, den, E);

    // 5. normalize + gather
    gat_out<<<NB, 128, 0, stream>>>(node_idx, num, den, (float*)d_out, NB);
}